// YOLOV3PostProcess_83021717832111
// MI455X (gfx1250) — compile-verified
//
#include <hip/hip_runtime.h>
#include <hip/hip_bf16.h>
#include <stdint.h>

#define NC 80
#define NTOT 22743
#define M_SCORES (NTOT * NC)      // 1,819,440
#define BATCH 8
#define PRE_NMS 400
#define TOPK 200
#define SCORE_THRESH 0.01f
#define NMS_THRESH 0.45f
#define PADN 416                  // 26 * 16
#define NWORDS 13                 // 416 bits / 32
#define SEL_THREADS 1024
#define CHUNK_FLOATS 4096         // 1024 threads * 4 floats (16B each)

// ---- WMMA availability (f32 16x16x4) ----
#ifndef HAVE_WMMA_F32
#  if defined(__has_builtin)
#    if __has_builtin(__builtin_amdgcn_wmma_f32_16x16x4_f32)
#      define HAVE_WMMA_F32 1
#    else
#      define HAVE_WMMA_F32 0
#    endif
#  else
#    define HAVE_WMMA_F32 0
#  endif
#endif

typedef __attribute__((ext_vector_type(2))) float v2f;
typedef __attribute__((ext_vector_type(8))) float v8f;

static __device__ __forceinline__ float sigm(float x) { return 1.0f / (1.0f + __expf(-x)); }

// monotone float->u32 key: larger float => larger unsigned key
static __device__ __forceinline__ uint32_t fkey(float f) {
  uint32_t u = __float_as_uint(f);
  return u ^ ((u & 0x80000000u) ? 0xFFFFFFFFu : 0x80000000u);
}
static __device__ __forceinline__ float funkey(uint32_t k) {
  uint32_t u = (k & 0x80000000u) ? (k ^ 0x80000000u) : ~k;
  return __uint_as_float(u);
}

// ---------------- Kernel 1: decode + score ----------------
__global__ void yolo_decode_kernel(const float* __restrict__ f0,
                                   const float* __restrict__ f1,
                                   const float* __restrict__ f2,
                                   float* __restrict__ boxes,
                                   float* __restrict__ scores) {
  int gid = blockIdx.x * blockDim.x + threadIdx.x;
  if (gid >= BATCH * NTOT) return;
  int b = gid / NTOT;
  int i = gid - b * NTOT;

  const float* fp;
  int Ww, hw, il;
  float st, aw, ah;
  if (i < 1083) {               // 19x19, stride 32
    fp = f0; Ww = 19; hw = 361; st = 32.f; il = i;
    int a = il / 361;
    const float AW[3] = {116.f, 156.f, 373.f}, AH[3] = {90.f, 198.f, 326.f};
    aw = AW[a]; ah = AH[a];
  } else if (i < 5415) {        // 38x38, stride 16
    fp = f1; Ww = 38; hw = 1444; st = 16.f; il = i - 1083;
    int a = il / 1444;
    const float AW[3] = {30.f, 62.f, 59.f}, AH[3] = {61.f, 45.f, 119.f};
    aw = AW[a]; ah = AH[a];
  } else {                      // 76x76, stride 8
    fp = f2; Ww = 76; hw = 5776; st = 8.f; il = i - 5415;
    int a = il / 5776;
    const float AW[3] = {10.f, 16.f, 33.f}, AH[3] = {13.f, 30.f, 23.f};
    aw = AW[a]; ah = AH[a];
  }
  int a = il / hw;
  int rem = il - a * hw;
  int y = rem / Ww;
  int x = rem - y * Ww;

  const float* base = fp + ((size_t)b * 255 + (size_t)a * 85) * (size_t)hw + (size_t)y * Ww + x;
  float t0 = base[0];
  float t1 = base[(size_t)hw];
  float t2 = base[(size_t)2 * hw];
  float t3 = base[(size_t)3 * hw];
  float t4 = base[(size_t)4 * hw];

  float bx = (sigm(t0) + (float)x) * st;
  float by = (sigm(t1) + (float)y) * st;
  float bw = __expf(t2) * aw;
  float bh = __expf(t3) * ah;
  float conf = sigm(t4);

  float* bo = boxes + ((size_t)b * NTOT + i) * 4;
  float4 bx4;
  bx4.x = bx - bw * 0.5f;
  bx4.y = by - bh * 0.5f;
  bx4.z = bx + bw * 0.5f;
  bx4.w = by + bh * 0.5f;
  *(float4*)bo = bx4;   // boxes region is 16B-aligned, stride 16B

  // scores region is 16B-aligned (i*NC*4 = i*320 bytes): vector stores
  float4* so4 = (float4*)(scores + (size_t)b * M_SCORES + (size_t)i * NC);
  for (int c4 = 0; c4 < NC / 4; c4++) {
    if ((c4 & 3) == 0 && (4 * c4 + 21) < NC)  // prefetch a later class-channel plane
      __builtin_prefetch(base + (size_t)(5 + 4 * c4 + 16) * hw, 0, 1);
    float4 sv;
    {
      float s0 = conf * sigm(base[(size_t)(5 + 4 * c4 + 0) * hw]);
      float s1 = conf * sigm(base[(size_t)(5 + 4 * c4 + 1) * hw]);
      float s2 = conf * sigm(base[(size_t)(5 + 4 * c4 + 2) * hw]);
      float s3 = conf * sigm(base[(size_t)(5 + 4 * c4 + 3) * hw]);
      sv.x = (s0 >= SCORE_THRESH) ? s0 : -1.0f;
      sv.y = (s1 >= SCORE_THRESH) ? s1 : -1.0f;
      sv.z = (s2 >= SCORE_THRESH) ? s2 : -1.0f;
      sv.w = (s3 >= SCORE_THRESH) ? s3 : -1.0f;
    }
    so4[c4] = sv;
  }
}

// ---------------- Kernel 2: exact top-400 radix select (async LDS staging) ----------------
__global__ __launch_bounds__(SEL_THREADS) void yolo_select_kernel(
    const float* __restrict__ scores,
    uint32_t* __restrict__ topkey,
    uint32_t* __restrict__ topidx) {
  const int b = blockIdx.x;
  const int t = threadIdx.x;
  const float* sc = scores + (size_t)b * M_SCORES;

  __shared__ float stage[2][CHUNK_FLOATS];           // 32 KB, async-staged
  __shared__ uint32_t hist[256];
  __shared__ uint32_t s_sel, s_krem, s_cnt;
  __shared__ unsigned long long buf[SEL_THREADS];    // 8 KB sort buffer

  const int NCH = M_SCORES / CHUNK_FLOATS;           // 444 full chunks
  const uint32_t lds0 = (uint32_t)(uintptr_t)&stage[0][t * 4];
  const uint32_t lds1 = (uint32_t)(uintptr_t)&stage[1][t * 4];

  uint32_t prefix = 0, prefmask = 0;
  int krem = PRE_NMS;

  for (int p = 0; p < 4; p++) {
    const int shift = 24 - 8 * p;
    for (int h = t; h < 256; h += SEL_THREADS) hist[h] = 0;
    __syncthreads();

    // prime chunk 0 into buffer 0 (each lane loads its own 16 bytes)
    {
      uint64_t g = (uint64_t)(uintptr_t)(sc + (size_t)t * 4);
      asm volatile("global_load_async_to_lds_b128 %0, %1, off" :: "v"(lds0), "v"(g) : "memory");
    }
    for (int c = 0; c < NCH; c++) {
      if (c + 1 < NCH) {
        uint64_t g = (uint64_t)(uintptr_t)(sc + (size_t)(c + 1) * CHUNK_FLOATS + (size_t)t * 4);
        uint32_t dst = (c & 1) ? lds0 : lds1;   // chunk c+1 -> buffer (c+1)&1
        asm volatile("global_load_async_to_lds_b128 %0, %1, off" :: "v"(dst), "v"(g) : "memory");
        asm volatile("s_wait_asynccnt 1" ::: "memory");
      } else {
        asm volatile("s_wait_asynccnt 0" ::: "memory");
      }
      const float* sp = &stage[c & 1][t * 4];   // each thread reads only its own lane's data
#pragma unroll
      for (int q = 0; q < 4; q++) {
        uint32_t k = fkey(sp[q]);
        if ((k & prefmask) == prefix) atomicAdd(&hist[(k >> shift) & 0xFFu], 1u);
      }
    }
    // tail (816 elements)
    for (int r = NCH * CHUNK_FLOATS + t; r < M_SCORES; r += SEL_THREADS) {
      uint32_t k = fkey(sc[r]);
      if ((k & prefmask) == prefix) atomicAdd(&hist[(k >> shift) & 0xFFu], 1u);
    }
    __syncthreads();
    if (t == 0) {
      uint32_t above = 0, selb = 0;
      for (int bkt = 255; bkt >= 0; bkt--) {
        uint32_t h = hist[bkt];
        if (above + h >= (uint32_t)krem) { selb = (uint32_t)bkt; break; }
        above += h;
      }
      s_sel = selb;
      s_krem = (uint32_t)krem - above;
    }
    __syncthreads();
    prefix |= (s_sel << shift);
    prefmask |= (0xFFu << shift);
    krem = (int)s_krem;
    __syncthreads();
  }

  const uint32_t pivot = prefix;  // exact value of the 400th-largest key
  if (t == 0) s_cnt = 0;
  __syncthreads();
  for (int r = t; r < M_SCORES; r += SEL_THREADS) {
    uint32_t k = fkey(sc[r]);
    if (k >= pivot) {
      uint32_t pos = atomicAdd(&s_cnt, 1u);
      if (pos < (uint32_t)SEL_THREADS)
        buf[pos] = ((unsigned long long)k << 32) | (uint32_t)(0xFFFFFFFFu - (uint32_t)r);
    }
  }
  __syncthreads();
  uint32_t cnt = s_cnt;
  if (cnt > SEL_THREADS) cnt = SEL_THREADS;
  if ((uint32_t)t >= cnt) buf[t] = 0ull;
  __syncthreads();

  // bitonic sort descending by (key desc, index asc) over 1024 u64
  for (int k2 = 2; k2 <= SEL_THREADS; k2 <<= 1) {
    for (int j = k2 >> 1; j > 0; j >>= 1) {
      int ixj = t ^ j;
      if (ixj > t) {
        unsigned long long A = buf[t], B = buf[ixj];
        bool desc = ((t & k2) == 0);
        if (desc ? (A < B) : (A > B)) { buf[t] = B; buf[ixj] = A; }
      }
      __syncthreads();
    }
  }

  for (int r = t; r < PRE_NMS; r += SEL_THREADS) {
    unsigned long long v = buf[r];
    uint32_t k = (uint32_t)(v >> 32);
    uint32_t idx = k ? (0xFFFFFFFFu - (uint32_t)v) : 0u;
    topkey[b * PRE_NMS + r] = k;
    topidx[b * PRE_NMS + r] = idx;
  }
}

// ---------------- Kernel 3: NMS + output (WMMA outer-sum for IoU union) ----------------
__global__ __launch_bounds__(512) void yolo_nms_kernel(
    const float* __restrict__ boxes,
    const uint32_t* __restrict__ topkey,
    const uint32_t* __restrict__ topidx,
    float* __restrict__ out) {
  const int b = blockIdx.x;
  const int t = threadIdx.x;

  __shared__ float X1[PADN], Y1[PADN], X2[PADN], Y2[PADN], AR[PADN], VAL[PADN];
  __shared__ int CLSs[PADN];
  __shared__ unsigned char KEEP[PADN], VALIDs[PADN];
  __shared__ uint32_t SUP[PADN * NWORDS];
  __shared__ int RANK[PRE_NMS];
  __shared__ int s_kept;

  for (int j = t; j < PADN; j += 512) {
    if (j < PRE_NMS) {
      uint32_t k = topkey[b * PRE_NMS + j];
      uint32_t idx = topidx[b * PRE_NMS + j];
      float v = k ? funkey(k) : -1.0f;
      int bi = (int)(idx / NC);
      int cc = (int)(idx - (uint32_t)bi * NC);
      const float* bo = boxes + ((size_t)b * NTOT + bi) * 4;
      float x1 = bo[0], y1 = bo[1], x2 = bo[2], y2 = bo[3];
      X1[j] = x1; Y1[j] = y1; X2[j] = x2; Y2[j] = y2;
      AR[j] = fmaxf(x2 - x1, 0.f) * fmaxf(y2 - y1, 0.f);
      VAL[j] = v; CLSs[j] = cc;
      VALIDs[j] = (v >= SCORE_THRESH) ? 1 : 0;
      KEEP[j] = 1;
    } else {
      X1[j] = Y1[j] = X2[j] = Y2[j] = 0.f;
      AR[j] = 0.f; VAL[j] = -1.f; CLSs[j] = -1 - j;
      VALIDs[j] = 0; KEEP[j] = 0;
    }
  }
  for (int w = t; w < PADN * NWORDS; w += 512) SUP[w] = 0u;
  __syncthreads();

  // Suppression-candidate matrix, 16x16 tiles per wave (wave32).
  const int wave = t >> 5, lane = t & 31;
  const int half = lane >> 4, l16 = lane & 15;
  for (int tile = wave; tile < 26 * 26; tile += 16) {
    int ti = tile / 26, tj = tile - (tile / 26) * 26;
#if HAVE_WMMA_F32
    // D = A(16x4) x B(4x16): A=[area_i, 1, 0, 0], B=[ones; area_j; 0; 0]
    // A layout: lanes 0-15 M=0..15, v0=K0, v1=K1 (lanes 16-31: K2,K3 -> zero)
    // B layout: v0 lanes0-15 = K0 row (ones), lanes16-31 = K1 row (area_j); v1 = 0
    v2f av, bv;
    av.x = half ? 0.f : AR[ti * 16 + l16];
    av.y = half ? 0.f : 1.0f;
    bv.x = half ? AR[tj * 16 + l16] : 1.0f;
    bv.y = 0.f;
    v8f cz = {};
    v8f d = __builtin_amdgcn_wmma_f32_16x16x4_f32(false, av, false, bv, (short)0, cz, false, false);
#endif
#pragma unroll
    for (int r = 0; r < 8; r++) {
      int i = ti * 16 + r + half * 8;   // C/D layout: M = r (+8 for lanes 16-31)
      int j = tj * 16 + l16;            // N = lane % 16
#if HAVE_WMMA_F32
      float sum = d[r];                 // area_i + area_j
#else
      float sum = AR[i] + AR[j];
#endif
      float xx1 = fmaxf(X1[i], X1[j]), yy1 = fmaxf(Y1[i], Y1[j]);
      float xx2 = fminf(X2[i], X2[j]), yy2 = fminf(Y2[i], Y2[j]);
      float inter = fmaxf(xx2 - xx1, 0.f) * fmaxf(yy2 - yy1, 0.f);
      // iou > T  <=>  inter > T * max(union, 1e-9)   (both sides positive)
      float uni = fmaxf(sum - inter, 1e-9f);
      bool sup = (inter > NMS_THRESH * uni) && (CLSs[i] == CLSs[j]) && (j > i) &&
                 (i < PRE_NMS) && (j < PRE_NMS);
      if (sup) atomicOr(&SUP[i * NWORDS + (j >> 5)], 1u << (j & 31));
    }
  }
  __syncthreads();

  // Sequential greedy suppression (matches reference fori_loop)
  for (int i = 0; i < PRE_NMS; i++) {
    if (KEEP[i] && VALIDs[i]) {
      int j = t;
      if (j < PADN && j > i && KEEP[j]) {
        if (SUP[i * NWORDS + (j >> 5)] & (1u << (j & 31))) KEEP[j] = 0;
      }
    }
    __syncthreads();
  }

  if (t == 0) {
    int cnt = 0;
    for (int i = 0; i < PRE_NMS; i++) {
      if (KEEP[i] && VALIDs[i]) RANK[i] = cnt++;
      else RANK[i] = -1;
    }
    s_kept = cnt;
  }
  __syncthreads();

  float* ob = out + (size_t)b * TOPK * 6;
  for (int i = t; i < PRE_NMS; i += 512) {
    int r = RANK[i];
    if (r >= 0 && r < TOPK) {
      ob[r * 6 + 0] = X1[i];
      ob[r * 6 + 1] = Y1[i];
      ob[r * 6 + 2] = X2[i];
      ob[r * 6 + 3] = Y2[i];
      ob[r * 6 + 4] = VAL[i];
      ob[r * 6 + 5] = (float)CLSs[i];
    }
  }
  int kept = s_kept;
  if (kept > TOPK) kept = TOPK;
  for (int r = kept + t; r < TOPK; r += 512) {
    ob[r * 6 + 0] = 0.f; ob[r * 6 + 1] = 0.f; ob[r * 6 + 2] = 0.f;
    ob[r * 6 + 3] = 0.f; ob[r * 6 + 4] = 0.f; ob[r * 6 + 5] = -1.f;
  }
}

// ---------------- launch ----------------
extern "C" void kernel_launch(void* const* d_in, const int* in_sizes, int n_in,
                              void* d_out, int out_size, void* d_ws, size_t ws_size,
                              hipStream_t stream) {
  (void)in_sizes; (void)n_in; (void)out_size; (void)ws_size;
  const float* f0 = (const float*)d_in[0];  // (8,255,19,19)
  const float* f1 = (const float*)d_in[1];  // (8,255,38,38)
  const float* f2 = (const float*)d_in[2];  // (8,255,76,76)
  float* out = (float*)d_out;               // (8,200,6)

  // workspace layout (16B-aligned segments): boxes | scores | topkey | topidx
  float* boxes = (float*)d_ws;                                   // 8*22743*4 floats
  float* scores = boxes + (size_t)BATCH * NTOT * 4;              // 8*1,819,440 floats (offset %16 == 0)
  uint32_t* topkey = (uint32_t*)(scores + (size_t)BATCH * M_SCORES);
  uint32_t* topidx = topkey + (size_t)BATCH * PRE_NMS;

  int tot = BATCH * NTOT;
  yolo_decode_kernel<<<(tot + 255) / 256, 256, 0, stream>>>(f0, f1, f2, boxes, scores);
  yolo_select_kernel<<<BATCH, SEL_THREADS, 0, stream>>>(scores, topkey, topidx);
  yolo_nms_kernel<<<BATCH, 512, 0, stream>>>(boxes, topkey, topidx, out);
}